// GCNLayer_52381421142408
// MI455X (gfx1250) — compile-verified
//
#include <hip/hip_runtime.h>

typedef __attribute__((ext_vector_type(2))) float v2f;
typedef __attribute__((ext_vector_type(8))) float v8f;

#define N_NODES 50000
#define N_EDGES 800000
#define K_DIM   512
#define H_DIM   512

#define LDS_ROW 516   // 512 + 4 pad floats -> ds_load_b64 A-reads hit all 64 banks once

// ---------------------------------------------------------------- zero out
__global__ __launch_bounds__(256) void gcn_zero(float4* __restrict__ out, int n4) {
    int i = blockIdx.x * 256 + threadIdx.x;
    if (i < n4) out[i] = make_float4(0.f, 0.f, 0.f, 0.f);
}

// ---------------------------------------------------------------- Z = X @ W (fp32 WMMA)
// Block (8 waves) computes a 16(M) x 256(N) slab. The 16x512 fp32 X slab
// (contiguous 32KB, rows are consecutive) is staged into LDS once per block
// via GLOBAL_LOAD_ASYNC_TO_LDS_B128 (ASYNCcnt), then each wave runs a
// 16x16x4 f32 WMMA K-loop with A from LDS and B (W, 1MB, L2-hot) from global.
__global__ __launch_bounds__(256) void gcn_gemm(const float* __restrict__ X,
                                                const float* __restrict__ W,
                                                float* __restrict__ Z) {
    __shared__ float lds[16 * LDS_ROW];          // 33,024 B

    const int tid  = threadIdx.x;
    const int lane = tid & 31;
    const int wave = tid >> 5;

    const int rowTile = blockIdx.x >> 1;         // 3125 row tiles of 16
    const int colBlk  = blockIdx.x & 1;          // 2 column half-slabs of 256
    const int row0 = rowTile * 16;
    const int col0 = colBlk * 256 + wave * 32;   // this wave's 32-wide tile

    const int laneM = lane & 15;                 // M for A, N for B/C
    const int half  = lane >> 4;                 // K-pair selector (ISA A-layout)

    // ---- async copy X slab -> LDS (2048 transfers of 16B; 8 per thread)
    {
        const unsigned long long gbase =
            (unsigned long long)(const char*)(X + (size_t)row0 * K_DIM);
        const unsigned lbase = (unsigned)(size_t)lds;
#pragma unroll
        for (int i = 0; i < 8; ++i) {
            const int flat = i * 256 + tid;      // 0..2047
            const int r    = flat >> 7;          // 128 x 16B chunks per row
            const int c    = flat & 127;
            const unsigned           ldsOff = lbase + (unsigned)(r * (LDS_ROW * 4) + c * 16);
            const unsigned long long gaddr  = gbase + (unsigned long long)(r * 2048 + c * 16);
            asm volatile("global_load_async_to_lds_b128 %0, %1, off"
                         :: "v"(ldsOff), "v"(gaddr)
                         : "memory");
        }
    }
#if __has_builtin(__builtin_amdgcn_s_wait_asynccnt)
    __builtin_amdgcn_s_wait_asynccnt(0);
#else
    asm volatile("s_wait_asynccnt 0x0" ::: "memory");
#endif
    __syncthreads();

    // ---- WMMA K-loop: A from LDS, B from global
    const float* __restrict__ arow = lds + laneM * LDS_ROW + 2 * half;

    v8f acc0 = {};
    v8f acc1 = {};

#pragma unroll 4
    for (int k = 0; k < K_DIM; k += 4) {
        v2f a = *(const v2f*)(arow + k);         // ds_load_b64, conflict-free

        const int kr = k + 2 * half;
        const float* __restrict__ wr = W + (size_t)kr * H_DIM + col0 + laneM;
        v2f b0, b1;
        b0.x = wr[0];
        b0.y = wr[H_DIM];
        b1.x = wr[16];
        b1.y = wr[H_DIM + 16];

        acc0 = __builtin_amdgcn_wmma_f32_16x16x4_f32(false, a, false, b0,
                                                     (short)0, acc0, false, false);
        acc1 = __builtin_amdgcn_wmma_f32_16x16x4_f32(false, a, false, b1,
                                                     (short)0, acc1, false, false);
    }

    // C/D layout: VGPR r -> M = r + 8*half, N = laneM (ISA 16x16 f32 table)
    float* __restrict__ zr = Z + (size_t)(row0 + 8 * half) * H_DIM + col0 + laneM;
#pragma unroll
    for (int r = 0; r < 8; ++r) {
        zr[(size_t)r * H_DIM]      = acc0[r];
        zr[(size_t)r * H_DIM + 16] = acc1[r];
    }
}

// ---------------------------------------------------------------- edge scatter
// One wave per edge: coalesced float4 gather of Z[src] (2KB row), scale by
// edge weight, fp32 atomic-add into out[dst] (output is L2-resident: 102MB).
__global__ __launch_bounds__(256) void gcn_scatter(const float* __restrict__ Z,
                                                   const int*   __restrict__ src,
                                                   const int*   __restrict__ dst,
                                                   const float* __restrict__ vals,
                                                   float* __restrict__ out,
                                                   int nEdges) {
    const int lane = threadIdx.x & 31;
    const int e    = (blockIdx.x * 256 + threadIdx.x) >> 5;
    if (e >= nEdges) return;

    const int   s  = src[e];
    const int   d  = dst[e];
    const float wv = vals[e];

    const float4* __restrict__ zrow = (const float4*)(Z + (size_t)s * H_DIM);
    float* __restrict__ orow        = out + (size_t)d * H_DIM;

#pragma unroll
    for (int c = 0; c < 4; ++c) {
        float4 v = zrow[lane + 32 * c];
        const int base = (lane + 32 * c) * 4;
        atomicAdd(orow + base + 0, wv * v.x);
        atomicAdd(orow + base + 1, wv * v.y);
        atomicAdd(orow + base + 2, wv * v.z);
        atomicAdd(orow + base + 3, wv * v.w);
    }
}

// ---------------------------------------------------------------- ReLU in place
__global__ __launch_bounds__(256) void gcn_relu(float4* __restrict__ out, int n4) {
    int i = blockIdx.x * 256 + threadIdx.x;
    if (i < n4) {
        float4 v = out[i];
        v.x = fmaxf(v.x, 0.f);
        v.y = fmaxf(v.y, 0.f);
        v.z = fmaxf(v.z, 0.f);
        v.w = fmaxf(v.w, 0.f);
        out[i] = v;
    }
}

// ---------------------------------------------------------------- launch
extern "C" void kernel_launch(void* const* d_in, const int* in_sizes, int n_in,
                              void* d_out, int out_size, void* d_ws, size_t ws_size,
                              hipStream_t stream) {
    const float* X     = (const float*)d_in[0];
    const float* W     = (const float*)d_in[1];
    const int*   esrc  = (const int*)d_in[2];
    const int*   edst  = (const int*)d_in[3];
    const float* evals = (const float*)d_in[4];
    float* out = (float*)d_out;
    float* Z   = (float*)d_ws;   // 50000*512*4 = 102.4 MB scratch

    const int n4 = N_NODES * H_DIM / 4;           // 6.4M float4

    gcn_zero<<<(n4 + 255) / 256, 256, 0, stream>>>((float4*)out, n4);

    const int gemmBlocks = (N_NODES / 16) * 2;    // 6250
    gcn_gemm<<<gemmBlocks, 256, 0, stream>>>(X, W, Z);

    const int scatBlocks = (N_EDGES * 32 + 255) / 256;   // 100000
    gcn_scatter<<<scatBlocks, 256, 0, stream>>>(Z, esrc, edst, evals, out, N_EDGES);

    gcn_relu<<<(n4 + 255) / 256, 256, 0, stream>>>((float4*)out, n4);
}